// DistMultPredictor_64501818851540
// MI455X (gfx1250) — compile-verified
//
#include <hip/hip_runtime.h>
#include <hip/hip_bf16.h>

typedef __attribute__((ext_vector_type(2))) float v2f;
typedef __attribute__((ext_vector_type(8))) float v8f;

#define N_DRUG   8000
#define N_DIS    18000
#define DIM      128
#define NREL     3
#define NEDGE    200000
#define TILES_PER_ROW   (NEDGE / 16)                   /* 12500 exact */
#define WAVES_PER_BLOCK 8
#define BLOCKS_PER_ROW  ((TILES_PER_ROW + WAVES_PER_BLOCK - 1) / WAVES_PER_BLOCK)

// One wave32 scores 16 edges of one relation-direction row.
// A-matrix (16x4 f32): lane m (m=lane&15) = edge e0+m; half-wave hi selects K pair.
// B-matrix = ones, so C accumulates per-edge dot products in every column.
// W row (block-uniform, 512 B) is staged in LDS so VMEM carries only the gathers.
__global__ __launch_bounds__(256) void distmult_wmma_kernel(
    const float* __restrict__ h_drug, const float* __restrict__ h_disease,
    const float* __restrict__ W,
    const int* __restrict__ drug_src, const int* __restrict__ dis_dst,
    const int* __restrict__ dis_src,  const int* __restrict__ drug_dst,
    float* __restrict__ out)
{
    __shared__ float wsh[DIM];

    const int r    = blockIdx.y;                 // 0..5 output row (uniform)
    const int wave = threadIdx.x >> 5;
    const int lane = threadIdx.x & 31;
    const int tile = blockIdx.x * WAVES_PER_BLOCK + wave;

    // Stage W row into LDS before any wave can exit (barrier needs all waves).
    if (threadIdx.x < DIM) wsh[threadIdx.x] = W[r * DIM + threadIdx.x];
    __syncthreads();

    if (tile >= TILES_PER_ROW) return;           // wave-uniform exit

    const int*   uidx;
    const int*   vidx;
    const float* utab;
    const float* vtab;
    if (r < NREL) {                               // forward: drug -> disease
        uidx = drug_src + r * NEDGE;  utab = h_drug;
        vidx = dis_dst  + r * NEDGE;  vtab = h_disease;
    } else {                                      // reverse: disease -> drug
        const int rr = r - NREL;
        uidx = dis_src  + rr * NEDGE; utab = h_disease;
        vidx = drug_dst + rr * NEDGE; vtab = h_drug;
    }
    float* orow = out + r * NEDGE;

    const int m  = lane & 15;                     // edge within tile
    const int hi = lane >> 4;                     // K-pair select
    const int e0 = tile * 16;

    const float* up = utab + (size_t)uidx[e0 + m] * DIM;
    const float* vp = vtab + (size_t)vidx[e0 + m] * DIM;

    v2f ones; ones.x = 1.0f; ones.y = 1.0f;
    v8f c = {};

    // 32 steps x K=4 covers all 128 dims; products in f32, reduction on matrix pipe.
#pragma unroll
    for (int s = 0; s < 32; ++s) {
        const int off = s * 4 + hi * 2;           // hi=0 -> K0,K1 ; hi=1 -> K2,K3
        v2f uu = *(const v2f*)(up  + off);
        v2f vv = *(const v2f*)(vp  + off);
        v2f ww = *(const v2f*)(&wsh[off]);        // ds_load_b64, conflict-free broadcast
        v2f a;
        a.x = uu.x * ww.x * vv.x;
        a.y = uu.y * ww.y * vv.y;
        c = __builtin_amdgcn_wmma_f32_16x16x4_f32(
                /*neg_a=*/false, a, /*neg_b=*/false, ones,
                /*c_mod=*/(short)0, c, /*reuse_a=*/false, /*reuse_b=*/false);
    }

    // C layout (32-bit 16x16): VGPR j holds row j (lanes 0-15) and row j+8 (lanes 16-31);
    // every column is identical, so c[j] = score[j] on lo half, score[j+8] on hi half.
#pragma unroll
    for (int j = 0; j < 8; ++j) {
        const float x   = c[j];
        const float sgm = 1.0f / (1.0f + __expf(-x));
        if (lane == j)      orow[e0 + j]     = sgm;   // lo half: edge j
        if (lane == 16 + j) orow[e0 + 8 + j] = sgm;   // hi half: edge j+8
    }
}

extern "C" void kernel_launch(void* const* d_in, const int* in_sizes, int n_in,
                              void* d_out, int out_size, void* d_ws, size_t ws_size,
                              hipStream_t stream) {
    const float* h_drug    = (const float*)d_in[0];
    const float* h_disease = (const float*)d_in[1];
    const float* W         = (const float*)d_in[2];
    const int*   drug_src  = (const int*)d_in[3];
    const int*   dis_dst   = (const int*)d_in[4];
    const int*   dis_src   = (const int*)d_in[5];
    const int*   drug_dst  = (const int*)d_in[6];
    float* out = (float*)d_out;

    dim3 grid(BLOCKS_PER_ROW, 2 * NREL);
    distmult_wmma_kernel<<<grid, 256, 0, stream>>>(
        h_drug, h_disease, W, drug_src, dis_dst, dis_src, drug_dst, out);
}